// GNN_Decode_80212809220210
// MI455X (gfx1250) — compile-verified
//
#include <hip/hip_runtime.h>
#include <hip/hip_bf16.h>
#include <math.h>

// ---------------------------------------------------------------------------
// GNN decode pipeline for MI455X (gfx1250), fp32 WMMA (V_WMMA_F32_16X16X4_F32)
// 256-thread blocks (8 wave32); weights streamed in 32x128 K-panels,
// double-buffered in LDS in K-pair-interleaved layout so each B-fragment is
// a single aligned ds_load_b64 (and pairs of fragments a ds_load_2addr_b64).
// ---------------------------------------------------------------------------

#define N_C 10000
#define E_C 160000
#define N_F 100000
#define E_F 800000
#define HDIM 128
#define BLK 256
#define WPB 8            // waves per block
#define PANEL_FLTS 4128  // 16 pair-rows * 129 float2 (padded pitch)

typedef float v2f __attribute__((ext_vector_type(2)));
typedef float v8f __attribute__((ext_vector_type(8)));

__device__ __forceinline__ v8f wmma_f32_16x16x4(v2f a, v2f b, v8f c) {
  return __builtin_amdgcn_wmma_f32_16x16x4_f32(false, a, false, b, (short)0, c,
                                               false, false);
}

// ---------------------------------------------------------------------------
// Multi-wave GEMM: each wave computes out[16x128] = act(in[16xK] @ W + bias).
// W ([Kx128] row-major, global) streamed in 32x128 panels. LDS panel layout:
// element (k,c) -> float2 slot [ (k>>1)*129 + c ], component (k&1).
// B frag for lane (c = nt*16 + r, k-pair = ks*2 + lane/16) = one float2.
// A frag (16x4 f32): lanes 0-15 K={k0,k0+1}, lanes 16-31 K={k0+2,k0+3}
// C/D   (16x16 f32): N = lane%16, M = vgpr + 8*(lane/16)
// ---------------------------------------------------------------------------
__device__ __forceinline__ void tile_gemm_multi(
    const float* sInTile, int ldIn, int K, const float* __restrict__ Wg,
    float* sW, const float* __restrict__ bias, bool act_elu, float* sOutTile,
    int ldOut, int lane, int tid) {
  const int r = lane & 15;
  const int half = lane >> 4;       // 0 or 1
  const int koff = half << 1;       // 0 or 2
  v8f acc[8];
  const v8f zero = {0.f, 0.f, 0.f, 0.f, 0.f, 0.f, 0.f, 0.f};
#pragma unroll
  for (int nt = 0; nt < 8; ++nt) acc[nt] = zero;

  // staging ownership: thread -> pair-row pp (k=2pp,2pp+1), col-quad c0..c0+7
  const int pp = tid >> 4;   // 0..15
  const int c0 = (tid & 15) * 8;
  const int P = K >> 5;      // K/32 panels
  float4 pA0, pA1, pB0, pB1; // rows 2pp / 2pp+1, cols c0..3 / c0+4..7
  {
    const float* g0 = Wg + (pp << 8) + c0;  // row 2pp of panel 0
    pA0 = ((const float4*)g0)[0];
    pA1 = ((const float4*)g0)[1];
    pB0 = ((const float4*)(g0 + HDIM))[0];
    pB1 = ((const float4*)(g0 + HDIM))[1];
  }
  for (int p = 0; p < P; ++p) {
    float2* buf2 = (float2*)(sW + (p & 1) * PANEL_FLTS);
    {
      float2* d = buf2 + pp * 129 + c0;
      d[0] = make_float2(pA0.x, pB0.x);
      d[1] = make_float2(pA0.y, pB0.y);
      d[2] = make_float2(pA0.z, pB0.z);
      d[3] = make_float2(pA0.w, pB0.w);
      d[4] = make_float2(pA1.x, pB1.x);
      d[5] = make_float2(pA1.y, pB1.y);
      d[6] = make_float2(pA1.z, pB1.z);
      d[7] = make_float2(pA1.w, pB1.w);
    }
    if (p + 1 < P) {
      const float* g0 = Wg + (size_t)(p + 1) * 4096 + (pp << 8) + c0;
      pA0 = ((const float4*)g0)[0];
      pA1 = ((const float4*)g0)[1];
      pB0 = ((const float4*)(g0 + HDIM))[0];
      pB1 = ((const float4*)(g0 + HDIM))[1];
    }
    __syncthreads();
#pragma unroll
    for (int ks = 0; ks < 8; ++ks) {
      const int k0 = (p << 5) + (ks << 2);
      v2f a;
      a.x = sInTile[r * ldIn + k0 + koff];
      a.y = sInTile[r * ldIn + k0 + koff + 1];
      const float2* w0 = buf2 + (ks * 2 + half) * 129 + r;
      float2 bf[8];
#pragma unroll
      for (int nt = 0; nt < 8; ++nt) bf[nt] = w0[nt * 16];
#pragma unroll
      for (int nt = 0; nt < 8; ++nt) {
        v2f b;
        b.x = bf[nt].x;
        b.y = bf[nt].y;
        acc[nt] = wmma_f32_16x16x4(a, b, acc[nt]);
      }
    }
    __syncthreads();
  }

  const int mb = half << 3;  // 0 or 8
#pragma unroll
  for (int nt = 0; nt < 8; ++nt) {
    const int cn = nt * 16 + r;
    const float bb = bias[cn];
#pragma unroll
    for (int v = 0; v < 8; ++v) {
      float val = acc[nt][v] + bb;
      if (act_elu) val = (val > 0.f) ? val : (__expf(val) - 1.f);
      sOutTile[(mb + v) * ldOut + cn] = val;
    }
  }
}

// Single-wave variant (decoder only; B streamed from global, weights tiny).
__device__ __forceinline__ void tile_gemm_16xK_128(
    const float* sIn, int ldIn, int K, const float* __restrict__ W,
    const float* __restrict__ bias, bool act_elu, float* sOut, int lane) {
  const int r = lane & 15;
  const int koff = (lane >> 4) << 1;
  v8f acc[8];
  const v8f zero = {0.f, 0.f, 0.f, 0.f, 0.f, 0.f, 0.f, 0.f};
#pragma unroll
  for (int nt = 0; nt < 8; ++nt) acc[nt] = zero;
  for (int k0 = 0; k0 < K; k0 += 4) {
    v2f a;
    a.x = sIn[r * ldIn + k0 + koff];
    a.y = sIn[r * ldIn + k0 + koff + 1];
    const float* w0 = W + (size_t)(k0 + koff) * HDIM + r;
#pragma unroll
    for (int nt = 0; nt < 8; ++nt) {
      v2f b;
      b.x = w0[nt * 16];
      b.y = w0[HDIM + nt * 16];
      acc[nt] = wmma_f32_16x16x4(a, b, acc[nt]);
    }
  }
  const int mb = (lane >> 4) << 3;
#pragma unroll
  for (int nt = 0; nt < 8; ++nt) {
    const int cn = nt * 16 + r;
    const float bb = bias[cn];
#pragma unroll
    for (int v = 0; v < 8; ++v) {
      float val = acc[nt][v] + bb;
      if (act_elu) val = (val > 0.f) ? val : (__expf(val) - 1.f);
      sOut[(mb + v) * HDIM + cn] = val;
    }
  }
}

// residual + LayerNorm over 128 feats; two lanes per row (64 cols each).
__device__ __forceinline__ void tile_ln_store(
    const float* sNew, int ldNew, const float* sRes, int ldR, int roff,
    const float* __restrict__ g, const float* __restrict__ b,
    float* __restrict__ out, int row0, int lane) {
  const int r = lane & 15;
  const int c0 = (lane >> 4) * 64;
  float part = 0.f;
  for (int j = 0; j < 64; ++j)
    part += sNew[r * ldNew + c0 + j] + sRes[r * ldR + roff + c0 + j];
  float tot = part + __shfl_xor(part, 16);
  float mu = tot * 0.0078125f;  // 1/128
  float vp = 0.f;
  for (int j = 0; j < 64; ++j) {
    float d = sNew[r * ldNew + c0 + j] + sRes[r * ldR + roff + c0 + j] - mu;
    vp += d * d;
  }
  float var = (vp + __shfl_xor(vp, 16)) * 0.0078125f;
  float inv = rsqrtf(var + 1e-5f);
  for (int j = 0; j < 64; ++j) {
    const int c = c0 + j;
    float d = sNew[r * ldNew + c] + sRes[r * ldR + roff + c] - mu;
    out[(size_t)(row0 + r) * HDIM + c] = d * inv * g[c] + b[c];
  }
}

// ---------------------------------------------------------------------------
// Edge update: e' = LN(e + MLP(cat(x_src, x_dst, e)))   (K = 384)
// 8 waves x 16 edges per block. Layer-2 output aliases sIn cols 0..127
// (x_src dead after layer 1); residual lives in cols 256..383.
// Dynamic LDS floats: 49152 (sIn) + 8256 (sW) + 16384 (sH) = 295168 B.
// ---------------------------------------------------------------------------
__global__ __launch_bounds__(BLK) void edge_mlp_kernel(
    const float* __restrict__ x, float* __restrict__ eattr,
    const int* __restrict__ eidx, int E, const float* __restrict__ W1,
    const float* __restrict__ b1, const float* __restrict__ W2,
    const float* __restrict__ b2, const float* __restrict__ lg,
    const float* __restrict__ lb) {
  extern __shared__ float smem[];
  float* sIn = smem;           // WPB * 16 * 384
  float* sW = smem + 49152;    // 2 * PANEL_FLTS
  float* sH = smem + 57408;    // WPB * 16 * 128
  __shared__ int sS[WPB * 16], sD[WPB * 16];
  const int tid = threadIdx.x;
  const int lane = tid & 31;
  const int w = tid >> 5;
  const int ntiles = E >> 4;
  int tile = blockIdx.x * WPB + w;
  const bool valid = tile < ntiles;
  if (!valid) tile = ntiles - 1;
  const int e0 = tile << 4;
  float* sInT = sIn + w * (16 * 384);
  float* sHT = sH + w * (16 * HDIM);
  __builtin_prefetch(W1, 0, 1);

  if (lane < 16) {
    sS[w * 16 + lane] = eidx[e0 + lane];
    sD[w * 16 + lane] = eidx[E + e0 + lane];
  }
  __syncthreads();
  for (int e = 0; e < 16; ++e) {
    float4 vs = ((const float4*)(x + (size_t)sS[w * 16 + e] * HDIM))[lane];
    float4 vd = ((const float4*)(x + (size_t)sD[w * 16 + e] * HDIM))[lane];
    float4 ve = ((const float4*)(eattr + (size_t)(e0 + e) * HDIM))[lane];
    ((float4*)(sInT + e * 384))[lane] = vs;
    ((float4*)(sInT + e * 384 + 128))[lane] = vd;
    ((float4*)(sInT + e * 384 + 256))[lane] = ve;
  }
  __syncthreads();
  tile_gemm_multi(sInT, 384, 384, W1, sW, b1, true, sHT, HDIM, lane, tid);
  __syncthreads();
  tile_gemm_multi(sHT, HDIM, HDIM, W2, sW, b2, false, sInT, 384, lane, tid);
  __syncthreads();
  if (valid) tile_ln_store(sInT, 384, sInT, 384, 256, lg, lb, eattr, e0, lane);
}

// ---------------------------------------------------------------------------
// Node update: x' = LN(x + MLP(cat(x, agg [, score*g_ae])))
// IN = 288 (coarse, attention) or 256 (fine). Layer-2 output aliases
// sIn cols 128..255 (agg dead after layer 1); residual = cols 0..127.
// Dynamic LDS floats: 36864 + 8256 + 16384 = 246016 B.
// ---------------------------------------------------------------------------
__global__ __launch_bounds__(BLK) void node_mlp_kernel(
    float* __restrict__ x, const float* __restrict__ agg,
    const float* __restrict__ cnt, int IN, const float* __restrict__ gae,
    const float* __restrict__ logit, const float* __restrict__ scal,
    const float* __restrict__ W1, const float* __restrict__ b1,
    const float* __restrict__ W2, const float* __restrict__ b2,
    const float* __restrict__ lg, const float* __restrict__ lb, int useAE) {
  extern __shared__ float smem[];
  float* sIn = smem;           // WPB * 16 * 288 (max)
  float* sW = smem + 36864;    // 2 * PANEL_FLTS
  float* sH = smem + 45120;    // WPB * 16 * 128
  const int tid = threadIdx.x;
  const int lane = tid & 31;
  const int w = tid >> 5;
  const int ntiles = (IN == 288 ? N_C : N_F) >> 4;
  int tile = blockIdx.x * WPB + w;
  const bool valid = tile < ntiles;
  if (!valid) tile = ntiles - 1;
  const int n0 = tile << 4;
  float* sInT = sIn + w * 16 * IN;
  float* sHT = sH + w * (16 * HDIM);
  __builtin_prefetch(W1, 0, 1);

  for (int e = 0; e < 16; ++e) {
    const int n = n0 + e;
    float4 vx = ((const float4*)(x + (size_t)n * HDIM))[lane];
    float4 va = ((const float4*)(agg + (size_t)n * HDIM))[lane];
    float c = fmaxf(cnt[n], 1.f);
    va.x /= c; va.y /= c; va.z /= c; va.w /= c;
    ((float4*)(sInT + e * IN))[lane] = vx;
    ((float4*)(sInT + e * IN + 128))[lane] = va;
    if (useAE) {
      float sc = __expf(logit[n] - scal[1]) / scal[2];  // softmax score
      sInT[e * IN + 256 + lane] = sc * gae[lane];
    }
  }
  __syncthreads();
  tile_gemm_multi(sInT, IN, IN, W1, sW, b1, true, sHT, HDIM, lane, tid);
  __syncthreads();
  tile_gemm_multi(sHT, HDIM, HDIM, W2, sW, b2, false, sInT + 128, IN, lane,
                  tid);
  __syncthreads();
  if (valid) tile_ln_store(sInT + 128, IN, sInT, IN, 0, lg, lb, x, n0, lane);
}

// ---------------------------------------------------------------------------
// Decoder: out = elu(x@W1+b1) @ W2[128x3] + b2   (single wave per 16 nodes)
// ---------------------------------------------------------------------------
__global__ __launch_bounds__(32) void decoder_kernel(
    const float* __restrict__ x, const float* __restrict__ W1,
    const float* __restrict__ b1, const float* __restrict__ W2,
    const float* __restrict__ b2, float* __restrict__ out) {
  __shared__ float sIn[16 * HDIM];
  __shared__ float sH[16 * HDIM];
  const int lane = threadIdx.x;
  const int n0 = blockIdx.x * 16;
  for (int e = 0; e < 16; ++e)
    ((float4*)(sIn + e * HDIM))[lane] =
        ((const float4*)(x + (size_t)(n0 + e) * HDIM))[lane];
  __syncthreads();
  tile_gemm_16xK_128(sIn, HDIM, HDIM, W1, b1, true, sH, lane);
  __syncthreads();
  const int r = lane & 15;
  const int c0 = (lane >> 4) * 64;
  float p0 = 0.f, p1 = 0.f, p2 = 0.f;
  for (int k = c0; k < c0 + 64; ++k) {
    float h = sH[r * HDIM + k];
    p0 += h * W2[k * 3 + 0];
    p1 += h * W2[k * 3 + 1];
    p2 += h * W2[k * 3 + 2];
  }
  p0 += __shfl_xor(p0, 16);
  p1 += __shfl_xor(p1, 16);
  p2 += __shfl_xor(p2, 16);
  if ((lane >> 4) == 0) {
    out[(size_t)(n0 + r) * 3 + 0] = p0 + b2[0];
    out[(size_t)(n0 + r) * 3 + 1] = p1 + b2[1];
    out[(size_t)(n0 + r) * 3 + 2] = p2 + b2[2];
  }
}

// ---------------------------------------------------------------------------
// Segment-mean pieces: scatter-add (fp32 atomics) + counts
// ---------------------------------------------------------------------------
__global__ void scatter_add_kernel(const float* __restrict__ eattr,
                                   const int* __restrict__ eidx, int E,
                                   float* __restrict__ agg,
                                   float* __restrict__ cnt) {
  long long gid = (long long)blockIdx.x * blockDim.x + threadIdx.x;
  if (gid >= (long long)E * 32) return;
  const int e = (int)(gid >> 5);
  const int q = (int)(gid & 31);
  const int dst = eidx[E + e];
  const float* s = eattr + (size_t)e * HDIM + q * 4;
  float* d = agg + (size_t)dst * HDIM + q * 4;
  atomicAdd(d + 0, s[0]);
  atomicAdd(d + 1, s[1]);
  atomicAdd(d + 2, s[2]);
  atomicAdd(d + 3, s[3]);
  if (q == 0) atomicAdd(cnt + dst, 1.f);
}

// ---------------------------------------------------------------------------
// Autoencoder: gpre[32] = x.flatten() @ W1   (W1 is [1.28M, 32] row-major)
// ---------------------------------------------------------------------------
__global__ void ae_partial_kernel(const float* __restrict__ xflat,
                                  const float* __restrict__ W,
                                  float* __restrict__ gpre, int NROWS) {
  __shared__ float red[256];
  const int t = threadIdx.x;
  const int j = t & 31;
  const int rg = t >> 5;
  float acc = 0.f;
  for (int i = blockIdx.x * 8 + rg; i < NROWS; i += gridDim.x * 8)
    acc += xflat[i] * W[(size_t)i * 32 + j];
  red[t] = acc;
  __syncthreads();
  if (rg == 0) {
    float s = red[j] + red[32 + j] + red[64 + j] + red[96 + j] + red[128 + j] +
              red[160 + j] + red[192 + j] + red[224 + j];
    atomicAdd(&gpre[j], s);
  }
}

// g_ae = elu(gpre + b1) @ W2[32x32] + b2 ; scal[0] = dot(g_ae, attw[128:160])
__global__ __launch_bounds__(32) void ae_finish_kernel(
    const float* __restrict__ gpre, const float* __restrict__ b1,
    const float* __restrict__ W2, const float* __restrict__ b2,
    const float* __restrict__ attw, float* __restrict__ gae,
    float* __restrict__ scal) {
  __shared__ float h[32];
  const int j = threadIdx.x;
  float v = gpre[j] + b1[j];
  v = (v > 0.f) ? v : (__expf(v) - 1.f);
  h[j] = v;
  __syncthreads();
  float o = b2[j];
  for (int k = 0; k < 32; ++k) o += h[k] * W2[k * 32 + j];
  gae[j] = o;
  float gd = o * attw[128 + j];
  for (int off = 16; off; off >>= 1) gd += __shfl_xor(gd, off);
  if (j == 0) scal[0] = gd;
}

// logit[n] = dot(x[n], attw[0:128]) + scal[0]   (warp per node)
__global__ void attn_logit_kernel(const float* __restrict__ x,
                                  const float* __restrict__ attw,
                                  const float* __restrict__ scal,
                                  float* __restrict__ logit, int N) {
  const int warp = threadIdx.x >> 5;
  const int lane = threadIdx.x & 31;
  const int n = blockIdx.x * 8 + warp;
  if (n >= N) return;
  float s = 0.f;
  for (int c = lane; c < HDIM; c += 32) s += x[(size_t)n * HDIM + c] * attw[c];
  for (int off = 16; off; off >>= 1) s += __shfl_xor(s, off);
  if (lane == 0) logit[n] = s + scal[0];
}

// single block: scal[1] = max(logit), scal[2] = sum(exp(logit - max))
__global__ void softmax_reduce_kernel(const float* __restrict__ logit, int N,
                                      float* __restrict__ scal) {
  __shared__ float red[1024];
  const int t = threadIdx.x;
  float m = -3.0e38f;
  for (int i = t; i < N; i += 1024) m = fmaxf(m, logit[i]);
  red[t] = m;
  __syncthreads();
  for (int s = 512; s; s >>= 1) {
    if (t < s) red[t] = fmaxf(red[t], red[t + s]);
    __syncthreads();
  }
  const float mx = red[0];
  __syncthreads();
  float se = 0.f;
  for (int i = t; i < N; i += 1024) se += __expf(logit[i] - mx);
  red[t] = se;
  __syncthreads();
  for (int s = 512; s; s >>= 1) {
    if (t < s) red[t] += red[t + s];
    __syncthreads();
  }
  if (t == 0) {
    scal[1] = mx;
    scal[2] = red[0];
  }
}

// unpool: dst[map[r]] = src[r] (128 floats per row, warp per row)
__global__ void scatter_rows_kernel(const float* __restrict__ src,
                                    const int* __restrict__ map,
                                    float* __restrict__ dst, int nrows) {
  const int row = (int)(((long long)blockIdx.x * blockDim.x + threadIdx.x) >> 5);
  const int lane = threadIdx.x & 31;
  if (row >= nrows) return;
  const int dr = map[row];
  ((float4*)(dst + (size_t)dr * HDIM))[lane] =
      ((const float4*)(src + (size_t)row * HDIM))[lane];
}

// ---------------------------------------------------------------------------
extern "C" void kernel_launch(void* const* d_in, const int* in_sizes, int n_in,
                              void* d_out, int out_size, void* d_ws,
                              size_t ws_size, hipStream_t stream) {
  (void)in_sizes; (void)n_in; (void)out_size; (void)ws_size;
  const float* x0 = (const float*)d_in[0];
  const float* ea0 = (const float*)d_in[1];
  const int* eiC = (const int*)d_in[2];
  const int* eiF = (const int*)d_in[3];
  const int* esel = (const int*)d_in[4];
  const int* perm = (const int*)d_in[5];
  auto F = [&](int i) { return (const float*)d_in[i]; };

  // dynamic LDS sizes (bytes)
  const int EDGE_SMEM = (49152 + 2 * PANEL_FLTS + 16384) * 4;  // 295168
  const int NODE_SMEM = (36864 + 2 * PANEL_FLTS + 16384) * 4;  // 246016
  hipFuncSetAttribute((const void*)edge_mlp_kernel,
                      hipFuncAttributeMaxDynamicSharedMemorySize, EDGE_SMEM);
  hipFuncSetAttribute((const void*)node_mlp_kernel,
                      hipFuncAttributeMaxDynamicSharedMemorySize, NODE_SMEM);

  // workspace layout (floats)
  float* ws = (float*)d_ws;
  float* xc = ws;                                  // N_C*H
  float* ec = xc + (size_t)N_C * HDIM;             // E_C*H
  float* xf = ec + (size_t)E_C * HDIM;             // N_F*H
  float* ef = xf + (size_t)N_F * HDIM;             // E_F*H
  float* agg = ef + (size_t)E_F * HDIM;            // N_F*H
  float* cnt = agg + (size_t)N_F * HDIM;           // N_F
  float* gpre = cnt + N_F;                         // 32
  float* gae = gpre + 32;                          // 32
  float* scal = gae + 32;                          // 4
  float* logit = scal + 4;                         // N_C

  hipMemcpyAsync(xc, x0, sizeof(float) * (size_t)N_C * HDIM,
                 hipMemcpyDeviceToDevice, stream);
  hipMemcpyAsync(ec, ea0, sizeof(float) * (size_t)E_C * HDIM,
                 hipMemcpyDeviceToDevice, stream);

  // ---- coarse level: 2 rounds with autoencoder + global attention ----
  for (int i = 0; i < 2; ++i) {
    edge_mlp_kernel<<<(E_C / 16 + WPB - 1) / WPB, BLK, EDGE_SMEM, stream>>>(
        xc, ec, eiC, E_C, F(6 + i * 2), F(14 + i * 2), F(7 + i * 2),
        F(15 + i * 2), F(22 + i), F(26 + i));
    hipMemsetAsync(agg, 0, sizeof(float) * (size_t)N_C * HDIM, stream);
    hipMemsetAsync(cnt, 0, sizeof(float) * N_C, stream);
    scatter_add_kernel<<<(E_C * 32 + 255) / 256, 256, 0, stream>>>(ec, eiC, E_C,
                                                                   agg, cnt);
    hipMemsetAsync(gpre, 0, sizeof(float) * 32, stream);
    ae_partial_kernel<<<512, 256, 0, stream>>>(xc, F(54 + i * 2), gpre,
                                               N_C * HDIM);
    ae_finish_kernel<<<1, 32, 0, stream>>>(gpre, F(58 + i * 2), F(55 + i * 2),
                                           F(59 + i * 2), F(62 + i), gae, scal);
    attn_logit_kernel<<<(N_C + 7) / 8, 256, 0, stream>>>(xc, F(62 + i), scal,
                                                         logit, N_C);
    softmax_reduce_kernel<<<1, 1024, 0, stream>>>(logit, N_C, scal);
    node_mlp_kernel<<<(N_C / 16 + WPB - 1) / WPB, BLK, NODE_SMEM, stream>>>(
        xc, agg, cnt, 288, gae, logit, scal, F(30 + i * 2), F(38 + i * 2),
        F(31 + i * 2), F(39 + i * 2), F(46 + i), F(50 + i), 1);
  }

  // ---- unpool: scatter coarse rows into zeroed fine buffers ----
  hipMemsetAsync(xf, 0, sizeof(float) * (size_t)N_F * HDIM, stream);
  hipMemsetAsync(ef, 0, sizeof(float) * (size_t)E_F * HDIM, stream);
  scatter_rows_kernel<<<(N_C * 32 + 255) / 256, 256, 0, stream>>>(xc, perm, xf,
                                                                  N_C);
  scatter_rows_kernel<<<(E_C * 32 + 255) / 256, 256, 0, stream>>>(ec, esel, ef,
                                                                  E_C);

  // ---- fine level: 2 rounds, no attention branch ----
  for (int i = 0; i < 2; ++i) {
    edge_mlp_kernel<<<(E_F / 16 + WPB - 1) / WPB, BLK, EDGE_SMEM, stream>>>(
        xf, ef, eiF, E_F, F(10 + i * 2), F(18 + i * 2), F(11 + i * 2),
        F(19 + i * 2), F(24 + i), F(28 + i));
    hipMemsetAsync(agg, 0, sizeof(float) * (size_t)N_F * HDIM, stream);
    hipMemsetAsync(cnt, 0, sizeof(float) * N_F, stream);
    scatter_add_kernel<<<(int)(((long long)E_F * 32 + 255) / 256), 256, 0,
                         stream>>>(ef, eiF, E_F, agg, cnt);
    node_mlp_kernel<<<(N_F / 16 + WPB - 1) / WPB, BLK, NODE_SMEM, stream>>>(
        xf, agg, cnt, 256, gae, logit, scal, F(34 + i * 2), F(42 + i * 2),
        F(35 + i * 2), F(43 + i * 2), F(48 + i), F(52 + i), 0);
  }

  // ---- decoder ----
  decoder_kernel<<<N_F / 16, 32, 0, stream>>>(xf, F(64), F(66), F(65), F(67),
                                              (float*)d_out);
}